// SchemaBank_4698694221848
// MI455X (gfx1250) — compile-verified
//
#include <hip/hip_runtime.h>
#include <math.h>

// ---------------- problem constants (from reference) ----------------
#define HDIM    2048   // hidden size H
#define SCHEMAS 32     // S
#define RANK    16     // r
#define AD      32     // attr dim
#define TILE_M  16     // WMMA M

typedef __attribute__((ext_vector_type(2))) float v2f;
typedef __attribute__((ext_vector_type(8))) float v8f;

// V_WMMA_F32_16X16X4_F32 : D(16x16 f32) = A(16x4 f32) * B(4x16 f32) + C
// 8-arg form: (neg_a, A, neg_b, B, c_mod, C, reuse_a, reuse_b)
__device__ __forceinline__ v8f wmma4(v2f a, v2f b, v8f c) {
  return __builtin_amdgcn_wmma_f32_16x16x4_f32(false, a, false, b, (short)0, c,
                                               false, false);
}

// ---------------- kernel 0: zero output + expert counters ----------------
__global__ __launch_bounds__(256) void zero_kernel(float* out, size_t n,
                                                   int* cnt) {
  size_t i = (size_t)blockIdx.x * 256 + threadIdx.x;
  size_t stride = (size_t)gridDim.x * 256;
  for (; i < n; i += stride) out[i] = 0.0f;
  if (blockIdx.x == 0 && threadIdx.x < SCHEMAS) cnt[threadIdx.x] = 0;
}

// ---------------- kernel 1: router (WMMA) + top-2 + list build ----------------
// One block = 16 tokens. 8 waves: wave w -> N-tile (w&1), K-chunk (w>>1)*512.
__global__ __launch_bounds__(256) void router_kernel(
    const float* __restrict__ h, const float* __restrict__ Wr,
    const float* __restrict__ attr, const float* __restrict__ Wa,
    int* __restrict__ cnt, int* __restrict__ tok, float* __restrict__ coef,
    int maxt) {
  __shared__ float part[8][16][16];   // per-wave partial score tiles
  __shared__ float sc[16][SCHEMAS];   // reduced router logits
  __shared__ float aw[SCHEMAS];       // sigmoid(attr @ Wa)

  const int tid = threadIdx.x;
  const int lane = tid & 31;
  const int wave = tid >> 5;

  // tiny attribute gate, computed once per block (wave 0)
  if (wave == 0) {
    float acc = 0.0f;
    const float* arow = attr + lane * AD;
#pragma unroll
    for (int j = 0; j < AD; ++j) acc += arow[j] * Wa[j];
    aw[lane] = 1.0f / (1.0f + __expf(-acc));
  }

  const int ntile = wave & 1;          // schemas [0..15] or [16..31]
  const int k0 = (wave >> 1) * 512;    // K chunk (2048/4 chunks of 512)
  const int hi = lane >> 4;            // lane-half selects K pair (ISA layout)
  const int m = lane & 15;             // A row (token) == B col index
  const int n = m + ntile * 16;        // schema column
  const int tg = blockIdx.x * TILE_M + m;
  const float* hrow = h + (size_t)tg * HDIM;
  const float* wrow = Wr + (size_t)n * HDIM;

  v8f c = {};
#pragma unroll 4
  for (int kk = 0; kk < 512; kk += 4) {
    const int k = k0 + kk + 2 * hi;
    v2f a = *(const v2f*)(hrow + k);   // A: lanes 0-15 K0/K1, 16-31 K2/K3
    v2f b = *(const v2f*)(wrow + k);   // B: same K striping, N across lanes
    c = wmma4(a, b, c);
  }
#pragma unroll
  for (int v = 0; v < 8; ++v) part[wave][v + 8 * hi][m] = c[v];
  __syncthreads();

  // deterministic fixed-order K reduction: sc[t][s] = sum_kc part[s16+2*kc]
  for (int i = tid; i < 16 * SCHEMAS; i += 256) {
    const int t = i >> 5;
    const int scol = i & 31;
    const int nt = scol >> 4;
    const int j = scol & 15;
    float acc = 0.0f;
#pragma unroll
    for (int kc = 0; kc < 4; ++kc) acc += part[nt + 2 * kc][t][j];
    sc[t][scol] = acc;
  }
  __syncthreads();

  // each wave post-processes 2 tokens; one lane per schema (wave32 == S)
  for (int tt = 0; tt < 2; ++tt) {
    const int t = wave * 2 + tt;
    const float val = sc[t][lane];

    float v1 = val; int i1 = lane;          // arg-top1 (ties -> lower index)
#pragma unroll
    for (int off = 16; off; off >>= 1) {
      float ov = __shfl_xor(v1, off, 32);
      int oi = __shfl_xor(i1, off, 32);
      if (ov > v1 || (ov == v1 && oi < i1)) { v1 = ov; i1 = oi; }
    }
    float v2 = (lane == i1) ? -3.4e38f : val; int i2 = lane;  // arg-top2
#pragma unroll
    for (int off = 16; off; off >>= 1) {
      float ov = __shfl_xor(v2, off, 32);
      int oi = __shfl_xor(i2, off, 32);
      if (ov > v2 || (ov == v2 && oi < i2)) { v2 = ov; i2 = oi; }
    }
    // full softmax then top-2 renorm (matches reference incl. 1e-8 eps)
    float e = __expf(val - v1);
    float Z = e;
#pragma unroll
    for (int off = 16; off; off >>= 1) Z += __shfl_xor(Z, off, 32);
    const float p1 = 1.0f / Z;
    const float p2 = __expf(v2 - v1) / Z;
    const float s2 = p1 + p2 + 1e-8f;
    const float g1 = p1 / s2, g2 = p2 / s2;
    const float scale = 0.9f + 0.2f * (g1 * aw[i1] + g2 * aw[i2]);

    if (lane == 0) {
      const int tok_g = blockIdx.x * TILE_M + t;
      int p = atomicAdd(&cnt[i1], 1);
      tok[i1 * maxt + p] = tok_g;
      coef[i1 * maxt + p] = scale * g1;
      p = atomicAdd(&cnt[i2], 1);
      tok[i2 * maxt + p] = tok_g;
      coef[i2 * maxt + p] = scale * g2;
    }
  }
}

// ---------------- kernel 3: per-expert low-rank GEMMs (WMMA) ----------------
// block = (expert s, chunk of 32 gathered tokens). 8 waves.
// GEMM1: X(32xH) @ U[s](Hx16) -> hU(32x16), K split over waves, fixed-order
//        reduce, coef folded. GEMM2: hU(32x16) @ V[s](16xH), atomics to out.
__global__ __launch_bounds__(256) void expert_kernel(
    const float* __restrict__ h, const float* __restrict__ U,
    const float* __restrict__ V, const int* __restrict__ cnt,
    const int* __restrict__ tok, const float* __restrict__ coef,
    float* __restrict__ out, int maxt) {
  __shared__ float part[8][32][16];
  __shared__ float hU[32][17];  // padded: conflict-free A reads in GEMM2
  __shared__ float cfs[32];
  __shared__ int tks[32];

  const int s = blockIdx.x;
  const int n0 = blockIdx.y * 32;
  const int cs = cnt[s];
  if (n0 >= cs) return;  // uniform per block

  const int tid = threadIdx.x;
  const int lane = tid & 31;
  const int wave = tid >> 5;

  if (tid < 32) {
    const int idx = n0 + tid;
    const bool valid = idx < cs;
    tks[tid] = tok[s * maxt + (valid ? idx : n0)];  // pad = dup token, cf 0
    cfs[tid] = valid ? coef[s * maxt + idx] : 0.0f;
  }
  __syncthreads();

  const int hi = lane >> 4;
  const int m = lane & 15;
  const float* Ub = U + (size_t)s * HDIM * RANK;
  const float* Vb = V + (size_t)s * RANK * HDIM;

  // ---- GEMM1: two 16-token M-tiles, K chunk of 256 per wave ----
  {
    const int k0 = wave * 256;
    const float* h0 = h + (size_t)tks[m] * HDIM;
    const float* h1 = h + (size_t)tks[m + 16] * HDIM;
    v8f c0 = {}, c1 = {};
#pragma unroll 4
    for (int kk = 0; kk < 256; kk += 4) {
      const int k = k0 + kk + 2 * hi;
      v2f a0 = *(const v2f*)(h0 + k);
      v2f a1 = *(const v2f*)(h1 + k);
      v2f b;
      b.x = Ub[k * RANK + m];            // B[k][r], r == lane&15
      b.y = Ub[(k + 1) * RANK + m];
      c0 = wmma4(a0, b, c0);
      c1 = wmma4(a1, b, c1);
    }
#pragma unroll
    for (int v = 0; v < 8; ++v) {
      part[wave][v + 8 * hi][m] = c0[v];
      part[wave][16 + v + 8 * hi][m] = c1[v];
    }
  }
  __syncthreads();

  // fixed-order wave reduction + fold gating coefficient (deterministic)
  for (int i = tid; i < 32 * RANK; i += 256) {
    const int M = i >> 4, r = i & 15;
    float acc = 0.0f;
#pragma unroll
    for (int w = 0; w < 8; ++w) acc += part[w][M][r];
    hU[M][r] = cfs[M] * acc;
  }
  __syncthreads();

  // ---- GEMM2: 128 N-tiles of 16 columns, 16 tiles per wave ----
  for (int nt = wave; nt < HDIM / 16; nt += 8) {
    const int col = nt * 16 + m;
    v8f c0 = {}, c1 = {};
#pragma unroll
    for (int kk = 0; kk < RANK; kk += 4) {
      const int k = kk + 2 * hi;
      v2f a0, a1, b;
      a0.x = hU[m][k];        a0.y = hU[m][k + 1];
      a1.x = hU[m + 16][k];   a1.y = hU[m + 16][k + 1];
      b.x = Vb[(size_t)k * HDIM + col];
      b.y = Vb[(size_t)(k + 1) * HDIM + col];
      c0 = wmma4(a0, b, c0);
      c1 = wmma4(a1, b, c1);
    }
#pragma unroll
    for (int v = 0; v < 8; ++v) {
      const int M = v + 8 * hi;
      __hip_atomic_fetch_add(&out[(size_t)tks[M] * HDIM + col], c0[v],
                             __ATOMIC_RELAXED, __HIP_MEMORY_SCOPE_AGENT);
      __hip_atomic_fetch_add(&out[(size_t)tks[M + 16] * HDIM + col], c1[v],
                             __ATOMIC_RELAXED, __HIP_MEMORY_SCOPE_AGENT);
    }
  }
}

// ---------------- host launcher ----------------
extern "C" void kernel_launch(void* const* d_in, const int* in_sizes, int n_in,
                              void* d_out, int out_size, void* d_ws,
                              size_t ws_size, hipStream_t stream) {
  const float* h    = (const float*)d_in[0];  // (B,T,H)
  const float* Wr   = (const float*)d_in[1];  // (S,H)
  const float* U    = (const float*)d_in[2];  // (S,H,r)
  const float* V    = (const float*)d_in[3];  // (S,r,H)
  const float* attr = (const float*)d_in[4];  // (S,ad)
  const float* Wa   = (const float*)d_in[5];  // (1,ad)
  float* out = (float*)d_out;

  const int BT = in_sizes[0] / HDIM;  // 8192 tokens
  const int maxt = BT;

  // workspace: [cnt: 32 ints (padded to 256B)] [tok: S*BT ints] [coef: S*BT f32]
  int* cnt = (int*)d_ws;
  int* tok = (int*)((char*)d_ws + 256);
  float* coef = (float*)((char*)d_ws + 256 + (size_t)SCHEMAS * maxt * sizeof(int));

  zero_kernel<<<2048, 256, 0, stream>>>(out, (size_t)out_size, cnt);
  router_kernel<<<BT / TILE_M, 256, 0, stream>>>(h, Wr, attr, Wa, cnt, tok,
                                                 coef, maxt);
  dim3 g3(SCHEMAS, (BT + 31) / 32);
  expert_kernel<<<g3, 256, 0, stream>>>(h, U, V, cnt, tok, coef, out, maxt);
}